// MambaBlock_3779571220937
// MI455X (gfx1250) — compile-verified
//
#include <hip/hip_runtime.h>
#include <math.h>

// ---------------------------------------------------------------------------
// Mamba block on gfx1250 (MI455X).
// Shapes: L=512, N_EMBD=192, D_INNER=384, DT_RANK=12, D_CONV=4, D_STATE=384.
// GEMMs use V_WMMA_F32_16X16X4_F32 (full fp32 precision, wave32, one 16x16
// tile per wave). The selective scan uses one wave per channel, 12 states per
// lane in VGPRs, hardware v_exp_f32, and wave32 shuffle reductions.
// ---------------------------------------------------------------------------

#define L_SEQ   512
#define N_EMBD  192
#define D_INNER 384
#define DT_RANK 12
#define D_CONV  4
#define D_STATE 384
#define DBC_N   (DT_RANK + 2 * D_STATE)   // 780

typedef float v2f __attribute__((ext_vector_type(2)));
typedef float v8f __attribute__((ext_vector_type(8)));

__device__ __forceinline__ v8f wmma4(v2f a, v2f b, v8f c) {
  // D = A(16x4) * B(4x16) + C(16x16), fp32 throughout.
  return __builtin_amdgcn_wmma_f32_16x16x4_f32(
      /*neg_a=*/false, a, /*neg_b=*/false, b,
      /*c_mod=*/(short)0, c, /*reuse_a=*/false, /*reuse_b=*/false);
}

__device__ __forceinline__ float silu_f(float x) {
  return x / (1.0f + __expf(-x));
}

// ---------------------------------------------------------------------------
// Kernel 1: xr = x @ W_in^T + b_in      (512 x 192) @ (192 x 768) -> (512x768)
// One wave per 16x16 tile. A lane holds A[m][k0..k1] / B^T rows analogously.
// ---------------------------------------------------------------------------
__global__ void k_gemm_in(const float* __restrict__ X,
                          const float* __restrict__ W,   // (768,192) row-major
                          const float* __restrict__ bias,
                          float* __restrict__ out) {
  const int lane = threadIdx.x & 31;
  const int wave = threadIdx.x >> 5;
  const int tile = blockIdx.x * (blockDim.x >> 5) + wave;
  const int TN = 768 / 16;                 // 48
  if (tile >= 32 * TN) return;
  const int tm = tile / TN, tn = tile % TN;
  const int row  = lane & 15;
  const int hi   = lane >> 4;
  const int koff = hi << 1;

  const float* arow = X + (tm * 16 + row) * N_EMBD;
  const float* brow = W + (tn * 16 + row) * N_EMBD;   // B[k][n] = W[n][k]

  v8f c = {};
  for (int k = 0; k < N_EMBD; k += 4) {
    v2f a = *(const v2f*)(arow + k + koff);
    v2f b = *(const v2f*)(brow + k + koff);
    c = wmma4(a, b, c);
  }
  const float bn = bias[tn * 16 + row];
#pragma unroll
  for (int r = 0; r < 8; ++r) {
    const int m = tm * 16 + r + 8 * hi;
    out[m * 768 + tn * 16 + row] = c[r] + bn;
  }
}

// ---------------------------------------------------------------------------
// Kernel 2: u = silu(depthwise_causal_conv(xr[:, :384])) , K=4
// ---------------------------------------------------------------------------
__global__ void k_conv_silu(const float* __restrict__ xr,
                            const float* __restrict__ cw,  // (384,1,4)
                            const float* __restrict__ cb,
                            float* __restrict__ u) {
  const int i = blockIdx.x * blockDim.x + threadIdx.x;
  if (i >= L_SEQ * D_INNER) return;
  const int l = i / D_INNER;
  const int d = i - l * D_INNER;
  float acc = cb[d];
#pragma unroll
  for (int k = 0; k < D_CONV; ++k) {
    const int t = l - (D_CONV - 1) + k;
    const float xi = (t >= 0) ? xr[t * 768 + d] : 0.0f;
    acc = fmaf(xi, cw[d * D_CONV + k], acc);
  }
  u[i] = silu_f(acc);
}

// ---------------------------------------------------------------------------
// Kernel 3: dbc = u @ W_xp^T          (512 x 384) @ (384 x 780) -> (512x780)
// N padded to 49 tiles of 16 (784); masked B loads keep EXEC uniform.
// ---------------------------------------------------------------------------
__global__ void k_gemm_xp(const float* __restrict__ U,
                          const float* __restrict__ W,   // (780,384)
                          float* __restrict__ dbc) {
  const int lane = threadIdx.x & 31;
  const int wave = threadIdx.x >> 5;
  const int tile = blockIdx.x * (blockDim.x >> 5) + wave;
  const int TN = 49;
  if (tile >= 32 * TN) return;
  const int tm = tile / TN, tn = tile % TN;
  const int row  = lane & 15;
  const int hi   = lane >> 4;
  const int koff = hi << 1;

  const int n_ = tn * 16 + row;
  const float msk = (n_ < DBC_N) ? 1.0f : 0.0f;
  const float* arow = U + (tm * 16 + row) * D_INNER;
  const float* brow = W + ((n_ < DBC_N) ? n_ : 0) * D_INNER;

  v8f c = {};
  for (int k = 0; k < D_INNER; k += 4) {
    v2f a = *(const v2f*)(arow + k + koff);
    v2f b = *(const v2f*)(brow + k + koff);
    b.x *= msk; b.y *= msk;
    c = wmma4(a, b, c);
  }
  if (n_ < DBC_N) {
#pragma unroll
    for (int r = 0; r < 8; ++r) {
      const int m = tm * 16 + r + 8 * hi;
      dbc[m * DBC_N + n_] = c[r];
    }
  }
}

// ---------------------------------------------------------------------------
// Kernel 4: delta = softplus(dbc[:, :12] @ W_dt^T)   K=12 -> 3 WMMA steps
// ---------------------------------------------------------------------------
__global__ void k_gemm_dt(const float* __restrict__ dbc,
                          const float* __restrict__ Wdt,  // (384,12)
                          float* __restrict__ delta) {
  const int lane = threadIdx.x & 31;
  const int wave = threadIdx.x >> 5;
  const int tile = blockIdx.x * (blockDim.x >> 5) + wave;
  const int TN = D_INNER / 16;             // 24
  if (tile >= 32 * TN) return;
  const int tm = tile / TN, tn = tile % TN;
  const int row  = lane & 15;
  const int hi   = lane >> 4;
  const int koff = hi << 1;

  const float* arow = dbc + (tm * 16 + row) * DBC_N;      // first 12 cols
  const float* brow = Wdt + (tn * 16 + row) * DT_RANK;

  v8f c = {};
#pragma unroll
  for (int k = 0; k < DT_RANK; k += 4) {
    v2f a = *(const v2f*)(arow + k + koff);
    v2f b = *(const v2f*)(brow + k + koff);
    c = wmma4(a, b, c);
  }
#pragma unroll
  for (int r = 0; r < 8; ++r) {
    const int m = tm * 16 + r + 8 * hi;
    const float v = c[r];
    // numerically stable softplus
    const float sp = (v > 20.0f) ? v : log1pf(expf(v));
    delta[m * D_INNER + tn * 16 + row] = sp;
  }
}

// ---------------------------------------------------------------------------
// Kernel 5: selective scan. One wave32 per channel d; each lane owns 12 of
// the 384 states in VGPRs. Per step: 12 exp + FMA updates, then a wave
// shuffle reduction for ys[l,d] = sum_n h[d,n] * C[l,n].
// ---------------------------------------------------------------------------
__global__ void k_scan(const float* __restrict__ delta,
                       const float* __restrict__ u,
                       const float* __restrict__ dbc,     // B at +12, C at +396
                       const float* __restrict__ A_log,
                       float* __restrict__ ys) {
  const int lane = threadIdx.x & 31;
  const int wave = threadIdx.x >> 5;
  const int d = blockIdx.x * (blockDim.x >> 5) + wave;
  if (d >= D_INNER) return;

  float An[12], h[12];
#pragma unroll
  for (int j = 0; j < 12; ++j) {
    An[j] = -__expf(A_log[d * D_STATE + lane + 32 * j]);
    h[j] = 0.0f;
  }

  for (int l = 0; l < L_SEQ; ++l) {
    const float dl = delta[l * D_INNER + d];   // broadcast load
    const float ul = u[l * D_INNER + d];
    const float du = dl * ul;
    const float* Brow = dbc + l * DBC_N + DT_RANK;
    const float* Crow = dbc + l * DBC_N + DT_RANK + D_STATE;
    float acc = 0.0f;
#pragma unroll
    for (int j = 0; j < 12; ++j) {
      const int n = lane + 32 * j;
      const float dA = __expf(dl * An[j]);
      h[j] = fmaf(dA, h[j], du * Brow[n]);
      acc = fmaf(h[j], Crow[n], acc);
    }
    // wave32 reduction
#pragma unroll
    for (int off = 16; off > 0; off >>= 1)
      acc += __shfl_xor(acc, off, 32);
    if (lane == 0) ys[l * D_INNER + d] = acc;
  }
}

// ---------------------------------------------------------------------------
// Kernel 6: y = (ys + u*D) * silu(res)   with res = xr[:, 384:768]
// ---------------------------------------------------------------------------
__global__ void k_gate(const float* __restrict__ ys,
                       const float* __restrict__ u,
                       const float* __restrict__ Dvec,
                       const float* __restrict__ xr,
                       float* __restrict__ y) {
  const int i = blockIdx.x * blockDim.x + threadIdx.x;
  if (i >= L_SEQ * D_INNER) return;
  const int l = i / D_INNER;
  const int d = i - l * D_INNER;
  const float r = xr[l * 768 + D_INNER + d];
  y[i] = fmaf(u[i], Dvec[d], ys[i]) * silu_f(r);
}

// ---------------------------------------------------------------------------
// Kernel 7: out = y @ W_out^T + b_out   (512 x 384) @ (384 x 192)
// ---------------------------------------------------------------------------
__global__ void k_gemm_out(const float* __restrict__ Y,
                           const float* __restrict__ W,   // (192,384)
                           const float* __restrict__ bias,
                           float* __restrict__ out) {
  const int lane = threadIdx.x & 31;
  const int wave = threadIdx.x >> 5;
  const int tile = blockIdx.x * (blockDim.x >> 5) + wave;
  const int TN = N_EMBD / 16;              // 12
  if (tile >= 32 * TN) return;
  const int tm = tile / TN, tn = tile % TN;
  const int row  = lane & 15;
  const int hi   = lane >> 4;
  const int koff = hi << 1;

  const float* arow = Y + (tm * 16 + row) * D_INNER;
  const float* brow = W + (tn * 16 + row) * D_INNER;

  v8f c = {};
  for (int k = 0; k < D_INNER; k += 4) {
    v2f a = *(const v2f*)(arow + k + koff);
    v2f b = *(const v2f*)(brow + k + koff);
    c = wmma4(a, b, c);
  }
  const float bn = bias[tn * 16 + row];
#pragma unroll
  for (int r = 0; r < 8; ++r) {
    const int m = tm * 16 + r + 8 * hi;
    out[m * N_EMBD + tn * 16 + row] = c[r] + bn;
  }
}

// ---------------------------------------------------------------------------
extern "C" void kernel_launch(void* const* d_in, const int* in_sizes, int n_in,
                              void* d_out, int out_size, void* d_ws, size_t ws_size,
                              hipStream_t stream) {
  const float* x      = (const float*)d_in[0];
  const float* W_in   = (const float*)d_in[1];
  const float* b_in   = (const float*)d_in[2];
  const float* conv_w = (const float*)d_in[3];
  const float* conv_b = (const float*)d_in[4];
  const float* W_xp   = (const float*)d_in[5];
  const float* W_dt   = (const float*)d_in[6];
  const float* A_log  = (const float*)d_in[7];
  const float* Dvec   = (const float*)d_in[8];
  const float* W_out  = (const float*)d_in[9];
  const float* b_out  = (const float*)d_in[10];
  float* out = (float*)d_out;

  // workspace layout (floats)
  float* ws    = (float*)d_ws;
  float* xr    = ws;                                  // 512*768
  float* u     = xr    + L_SEQ * 768;                 // 512*384
  float* dbc   = u     + L_SEQ * D_INNER;             // 512*780
  float* delta = dbc   + L_SEQ * DBC_N;               // 512*384
  float* ys    = delta + L_SEQ * D_INNER;             // 512*384
  float* y     = ys    + L_SEQ * D_INNER;             // 512*384

  const int WPB = 4;               // waves per block
  const int TPB = WPB * 32;        // 128 threads

  // 1) in-projection: 32 x 48 tiles
  k_gemm_in<<<(32 * 48) / WPB, TPB, 0, stream>>>(x, W_in, b_in, xr);

  // 2) conv + silu
  k_conv_silu<<<(L_SEQ * D_INNER + 255) / 256, 256, 0, stream>>>(xr, conv_w, conv_b, u);

  // 3) x-projection: 32 x 49 tiles (N padded 780 -> 784)
  k_gemm_xp<<<(32 * 49 + WPB - 1) / WPB, TPB, 0, stream>>>(u, W_xp, dbc);

  // 4) delta GEMM + softplus: 32 x 24 tiles
  k_gemm_dt<<<(32 * 24) / WPB, TPB, 0, stream>>>(dbc, W_dt, delta);

  // 5) selective scan: one wave per channel, 8 waves (256 thr) per block
  k_scan<<<D_INNER / 8, 256, 0, stream>>>(delta, u, dbc, A_log, ys);

  // 6) gating
  k_gate<<<(L_SEQ * D_INNER + 255) / 256, 256, 0, stream>>>(ys, u, Dvec, xr, y);

  // 7) out-projection: 32 x 12 tiles
  k_gemm_out<<<(32 * 12) / WPB, TPB, 0, stream>>>(y, W_out, b_out, out);
}